// CrossRankCriterion_46033459479118
// MI455X (gfx1250) — compile-verified
//
#include <hip/hip_runtime.h>
#include <math.h>

// NDCG@10 loss: B=128 rows, N=1000 docs/row, k=10.
// One single-wave (32-lane) workgroup per row: barriers degenerate to S_NOP
// (CDNA5 ISA 3.1), all combining via wave32 shuffles. Row staged into LDS with
// CDNA5 GLOBAL_LOAD_ASYNC_TO_LDS_B128 (ASYNCcnt). DCG: 10 rounds of stable
// wave argmax on u64-packed keys; IDCG: 5-bin label histogram (labels 0..4).

#define B_Q   128
#define N_D   1000
#define K_TOP 10
#define NPAD  1024
#define TPB   32      // one wave32 per row
#define SUMT  128

typedef int v4i __attribute__((ext_vector_type(4)));
typedef __attribute__((address_space(1))) v4i gbl_v4i;
typedef __attribute__((address_space(3))) v4i lds_v4i;

__device__ __forceinline__ unsigned long long pack_key(float v, int idx) {
    unsigned u = __float_as_uint(v);
    // monotone map: larger float -> larger unsigned
    u = (u & 0x80000000u) ? ~u : (u | 0x80000000u);
    // tie-break: among equal values prefer SMALLER index (stable argsort desc)
    return ((unsigned long long)u << 32) | (unsigned)(1023 - idx);
}

__global__ __launch_bounds__(TPB)
void ndcg_row_kernel(const float* __restrict__ preds,
                     const float* __restrict__ labels,
                     float* __restrict__ row_loss) {
    __shared__ float s_pred[NPAD];
    __shared__ float s_lab[NPAD];
    __shared__ int s_hist[5];

    const int row = blockIdx.x;
    const int tid = threadIdx.x;   // 0..31

    const float4* gp = (const float4*)(preds  + (size_t)row * N_D);
    const float4* gl = (const float4*)(labels + (size_t)row * N_D);

    // ---- Stage row into LDS: async tensor path on gfx1250 ----
#if __has_builtin(__builtin_amdgcn_global_load_async_to_lds_b128)
    for (int c = tid; c < N_D / 4; c += TPB) {   // 250 x 16B chunks
        __builtin_amdgcn_global_load_async_to_lds_b128(
            (gbl_v4i*)(gp + c), (lds_v4i*)&s_pred[c * 4], 0, 0);
        __builtin_amdgcn_global_load_async_to_lds_b128(
            (gbl_v4i*)(gl + c), (lds_v4i*)&s_lab[c * 4], 0, 0);
    }
#else
    for (int c = tid; c < N_D / 4; c += TPB) {
        ((float4*)s_pred)[c] = gp[c];
        ((float4*)s_lab)[c]  = gl[c];
    }
#endif

    // pad slots 1000..1023 so the strided max never picks them
    if (tid < NPAD - N_D) s_pred[N_D + tid] = -INFINITY;
    if (tid < 5) s_hist[tid] = 0;

#if __has_builtin(__builtin_amdgcn_s_wait_asynccnt)
    __builtin_amdgcn_s_wait_asynccnt(0);
#else
    asm volatile("s_wait_asynccnt 0" ::: "memory");
#endif
    __syncthreads();   // single-wave workgroup: S_NOP + dscnt ordering

    // ---- Label histogram (labels are exact floats 0..4) ----
    for (int i = tid; i < N_D; i += TPB) {
        atomicAdd(&s_hist[(int)s_lab[i]], 1);
    }
    __syncthreads();

    // denoms[i] = log2(i + 2), fp32
    const float denom[K_TOP] = {1.0f, 1.5849625f, 2.0f, 2.3219281f, 2.5849625f,
                                2.8073549f, 3.0f, 3.1699250f, 3.3219281f, 3.4594316f};

    // ---- DCG: 10 rounds of stable wave-wide argmax over LDS row ----
    float dcg = 0.0f;
    for (int it = 0; it < K_TOP; ++it) {
        unsigned long long key = 0ull;
        // 32 conflict-free LDS reads per lane (stride-32 over 64 banks)
        for (int i = tid; i < NPAD; i += TPB) {
            unsigned long long k = pack_key(s_pred[i], i);
            key = (k > key) ? k : key;
        }
        // wave32 butterfly max (no barriers, no LDS round-trip)
        #pragma unroll
        for (int m = 16; m > 0; m >>= 1) {
            unsigned long long o = __shfl_xor(key, m, TPB);
            key = (o > key) ? o : key;
        }
        const int idx = 1023 - (int)(unsigned)(key & 0xFFFFFFFFull);
        if (tid == 0) {
            float rel = exp2f(s_lab[idx]) - 1.0f;   // 2^label - 1 (exact)
            dcg += rel / denom[it];
            s_pred[idx] = -INFINITY;                // knock out winner
        }
        __syncthreads();   // order lane-0 knockout before next scan (dscnt)
    }

    // ---- IDCG from histogram + row loss ----
    if (tid == 0) {
        float idcg = 0.0f;
        int pos = 0;
        for (int v = 4; v >= 1; --v) {
            int c = s_hist[v];
            const float relv = (float)((1 << v) - 1);   // 15, 7, 3, 1
            while (c-- > 0 && pos < K_TOP) {
                idcg += relv / denom[pos];
                ++pos;
            }
        }
        row_loss[row] = 1.0f - dcg / idcg;
    }
}

// Deterministic fixed-order reduction of the 128 row losses.
__global__ __launch_bounds__(SUMT)
void ndcg_sum_kernel(const float* __restrict__ row_loss, float* __restrict__ out) {
    __shared__ float s[SUMT];
    const int tid = threadIdx.x;
    s[tid] = row_loss[tid];            // B_Q == SUMT == 128
    __syncthreads();
    for (int off = SUMT / 2; off > 0; off >>= 1) {
        if (tid < off) s[tid] += s[tid + off];
        __syncthreads();
    }
    if (tid == 0) out[0] = s[0];
}

extern "C" void kernel_launch(void* const* d_in, const int* in_sizes, int n_in,
                              void* d_out, int out_size, void* d_ws, size_t ws_size,
                              hipStream_t stream) {
    (void)in_sizes; (void)n_in; (void)out_size; (void)ws_size;
    const float* preds  = (const float*)d_in[0];
    const float* labels = (const float*)d_in[1];
    float* row_loss = (float*)d_ws;    // 128 floats of scratch

    ndcg_row_kernel<<<B_Q, TPB, 0, stream>>>(preds, labels, row_loss);
    ndcg_sum_kernel<<<1, SUMT, 0, stream>>>(row_loss, (float*)d_out);
}